// EruSelfAttentionModel_76544907149381
// MI455X (gfx1250) — compile-verified
//
#include <hip/hip_runtime.h>

// ---------------------------------------------------------------------------
// EruSelfAttention, MI455X (gfx1250, wave32) — full-f32 version.
// Algebraic collapse (output depends only on the last sequence position):
//   scores_last[b,h,t] = ((W_key[h]^T (W_query[h] e_last[b])) / sqrt(E)) . e[b,t]
//   sel = max(norm) == 1.0 exactly (IEEE x/x==1) -> softmax branch weight 0
//   ov_last[b,h]       = W_value[h] @ (sum_t w[b,h,t] * e[b,t])
// The two S-length contractions (M=8 padded to 16) run on the matrix pipe via
// V_WMMA_F32_16X16X4_F32 (f32 in / f32 acc -> matches the f32 reference).
// Workload is ~35 MMAC + 8 MB traffic: precision is free at 23.3 TB/s.
// 4 independent accumulator chains per wave hide the WMMA D->C RAW latency.
// ---------------------------------------------------------------------------

typedef __attribute__((ext_vector_type(2))) float v2f;
typedef __attribute__((ext_vector_type(8))) float v8f;

#define B_   2
#define S_   2048
#define E_   512
#define A_   64
#define H_   8
#define HPAD 16
#define INV_SCALE 0.04419417382415922f  // 1/sqrt(512)
#define SIG_DOM_  10.0f

// 32-bit A 16x4 fragment (ISA §7.12.2): lane L (row/col = L&15):
//   VGPR0 = K (lanes 0-15: k0+0, lanes 16-31: k0+2), VGPR1 = next K.
// -> one contiguous float2 per lane. B 4x16 mirrors it (col = L&15).
__device__ __forceinline__ v2f ld2(const float* __restrict__ row, int k0, bool lo) {
  return *(const v2f*)(row + k0 + (lo ? 0 : 2));
}

__device__ __forceinline__ v8f wmma4(v2f a, v2f b, v8f c) {
  return __builtin_amdgcn_wmma_f32_16x16x4_f32(false, a, false, b,
                                               (short)0, c, false, false);
}

// ---------------- kernel 1: embedding gather + LayerNorm (f32) -------------
__global__ void k_embed_ln(const int* __restrict__ x, const float* __restrict__ emb,
                           const float* __restrict__ gamma, const float* __restrict__ beta,
                           float* __restrict__ e, float* __restrict__ e_last) {
  const int tok = blockIdx.x;            // B*S blocks
  const int b = tok / S_, t = tok % S_;
  const int tid = threadIdx.x;           // 256 threads, 2 elems each
  __shared__ float red[256];

  const float* src = emb + (size_t)x[tok] * E_;
  float v0 = src[tid], v1 = src[tid + 256];

  red[tid] = v0 + v1;
  __syncthreads();
  for (int s = 128; s > 0; s >>= 1) { if (tid < s) red[tid] += red[tid + s]; __syncthreads(); }
  const float mu = red[0] * (1.0f / (float)E_);
  __syncthreads();

  const float d0 = v0 - mu, d1 = v1 - mu;
  red[tid] = d0 * d0 + d1 * d1;
  __syncthreads();
  for (int s = 128; s > 0; s >>= 1) { if (tid < s) red[tid] += red[tid + s]; __syncthreads(); }
  const float rstd = rsqrtf(red[0] * (1.0f / (float)E_) + 1e-5f);

  const float o0 = d0 * rstd * gamma[tid]       + beta[tid];
  const float o1 = d1 * rstd * gamma[tid + 256] + beta[tid + 256];

  float* er = e + (size_t)tok * E_;
  er[tid]       = o0;
  er[tid + 256] = o1;
  if (t == S_ - 1) {
    float* el = e_last + b * E_;
    el[tid] = o0; el[tid + 256] = o1;
  }
}

// ---------------- kernel 2: q_last + folded key vector r (f32) -------------
__global__ void k_qr(const float* __restrict__ Wq, const float* __restrict__ Wk,
                     const float* __restrict__ e_last, float* __restrict__ rpad) {
  const int bh = blockIdx.x;             // B*16 blocks
  const int b = bh / HPAD, h = bh % HPAD;
  const int tid = threadIdx.x;           // 64 threads
  float* rrow = rpad + (size_t)bh * E_;
  if (h >= H_) {                         // zero padding rows for WMMA
    for (int i = tid; i < E_; i += 64) rrow[i] = 0.0f;
    return;
  }
  __shared__ float se[E_];
  __shared__ float sq[A_];
  for (int i = tid; i < E_; i += 64) se[i] = e_last[b * E_ + i];
  __syncthreads();

  const float* wq = Wq + ((size_t)h * A_ + tid) * E_;
  float acc = 0.0f;
  for (int d = 0; d < E_; ++d) acc += wq[d] * se[d];
  sq[tid] = acc;
  __syncthreads();

  for (int i = 0; i < 8; ++i) {
    const int d = tid + i * 64;
    float r = 0.0f;
    for (int a = 0; a < A_; ++a) r += Wk[((size_t)h * A_ + a) * E_ + d] * sq[a];
    rrow[d] = r * INV_SCALE;             // fold 1/sqrt(E) into r
  }
}

// ---------------- kernel 3: scores_last = rpad @ e^T (f32 WMMA) ------------
// One wave per 16-token tile; K over E in 4 interleaved accumulator chains.
__global__ void k_scores(const float* __restrict__ rpad, const float* __restrict__ e,
                         float* __restrict__ scores) {
  const int blk = blockIdx.x;            // B * S/16 blocks, 32 threads each
  const int b  = blk / (S_ / 16);
  const int n0 = (blk % (S_ / 16)) * 16;
  const int lane = threadIdx.x;
  const bool lo = lane < 16;
  const int ln = lane & 15;
  const float* arow = rpad + ((size_t)b * HPAD + ln) * E_;     // A row M=ln
  const float* brow = e + ((size_t)b * S_ + (n0 + ln)) * E_;   // B col t=n0+ln
  v8f a0 = {}, a1 = {}, a2 = {}, a3 = {};
  for (int k0 = 0; k0 < E_; k0 += 16) {
    if (k0 + 16 < E_) __builtin_prefetch(brow + k0 + 16, 0, 0);
    a0 = wmma4(ld2(arow, k0,      lo), ld2(brow, k0,      lo), a0);
    a1 = wmma4(ld2(arow, k0 + 4,  lo), ld2(brow, k0 + 4,  lo), a1);
    a2 = wmma4(ld2(arow, k0 + 8,  lo), ld2(brow, k0 + 8,  lo), a2);
    a3 = wmma4(ld2(arow, k0 + 12, lo), ld2(brow, k0 + 12, lo), a3);
  }
  const v8f acc = (a0 + a1) + (a2 + a3);
  if (lo) {                              // rows 0..7 (= heads) in lanes 0..15
#pragma unroll
    for (int i = 0; i < H_; ++i)
      scores[((size_t)b * H_ + i) * S_ + n0 + ln] = acc[i];
  }
}

// ---------------- kernel 4: min/max -> sigmoid weights (f32) ---------------
__global__ void k_weights(const float* __restrict__ scores, float* __restrict__ wpad) {
  const int bh = blockIdx.x;             // B*16 blocks
  const int b = bh / HPAD, h = bh % HPAD;
  const int tid = threadIdx.x;           // 256 threads, 8 elems each
  float* wrow = wpad + (size_t)bh * S_;
  if (h >= H_) {
    for (int i = tid; i < S_; i += 256) wrow[i] = 0.0f;
    return;
  }
  const float* srow = scores + ((size_t)b * H_ + h) * S_;
  float v[8];
  float mx = -INFINITY, mn = INFINITY;
#pragma unroll
  for (int i = 0; i < 8; ++i) {
    v[i] = srow[tid + i * 256];
    mx = fmaxf(mx, v[i]); mn = fminf(mn, v[i]);
  }
  __shared__ float rmax[256], rmin[256];
  rmax[tid] = mx; rmin[tid] = mn;
  __syncthreads();
  for (int s = 128; s > 0; s >>= 1) {
    if (tid < s) {
      rmax[tid] = fmaxf(rmax[tid], rmax[tid + s]);
      rmin[tid] = fminf(rmin[tid], rmin[tid + s]);
    }
    __syncthreads();
  }
  mx = rmax[0]; mn = rmin[0];
  const float inv = 1.0f / (mx - mn);
  // sel == 1.0 exactly => weights are pure sigmoid of the normalized scores.
#pragma unroll
  for (int i = 0; i < 8; ++i) {
    const float norm = (v[i] - mn) * inv;
    const float z = 2.0f * SIG_DOM_ * norm - SIG_DOM_;
    wrow[tid + i * 256] = 1.0f / (1.0f + __expf(-z));
  }
}

// ---------------- kernel 5: u = wpad @ e (f32 WMMA over K=S) ---------------
// B[k=t][n=d] is a column access of e: at K-step 4 that is just two strided
// b32 loads per lane, so no transposed copy of e is needed.
__global__ void k_wsum(const float* __restrict__ wpad, const float* __restrict__ e,
                       float* __restrict__ u) {
  const int blk = blockIdx.x;            // B * E/16 blocks, 32 threads each
  const int b  = blk / (E_ / 16);
  const int n0 = (blk % (E_ / 16)) * 16;
  const int lane = threadIdx.x;
  const bool lo = lane < 16;
  const int ln = lane & 15;
  const float* arow = wpad + ((size_t)b * HPAD + ln) * S_;     // A row M=ln
  const float* ecol = e + (size_t)b * S_ * E_ + (n0 + ln);     // B col d=n0+ln
  v8f a0 = {}, a1 = {}, a2 = {}, a3 = {};
  for (int k0 = 0; k0 < S_; k0 += 16) {
    if (k0 + 16 < S_) {
      __builtin_prefetch(arow + k0 + 16, 0, 0);
      __builtin_prefetch(ecol + (size_t)(k0 + 16 + (lo ? 0 : 2)) * E_, 0, 0);
    }
#pragma unroll
    for (int j = 0; j < 4; ++j) {
      const int c = k0 + 4 * j + (lo ? 0 : 2);
      v2f bb;
      bb[0] = ecol[(size_t)c * E_];
      bb[1] = ecol[(size_t)(c + 1) * E_];
      const v2f aa = ld2(arow, k0 + 4 * j, lo);
      if (j == 0)      a0 = wmma4(aa, bb, a0);
      else if (j == 1) a1 = wmma4(aa, bb, a1);
      else if (j == 2) a2 = wmma4(aa, bb, a2);
      else             a3 = wmma4(aa, bb, a3);
    }
  }
  const v8f acc = (a0 + a1) + (a2 + a3);
  if (lo) {
#pragma unroll
    for (int i = 0; i < H_; ++i)
      u[((size_t)b * H_ + i) * E_ + n0 + ln] = acc[i];
  }
}

// ---------------- kernel 6: ov_last = W_value[h] @ u[b,h] ------------------
__global__ void k_ov(const float* __restrict__ Wv, const float* __restrict__ u,
                     float* __restrict__ ov) {
  const int bh = blockIdx.x;             // B*H blocks
  const int h = bh % H_;
  const int tid = threadIdx.x;           // 256 threads, 2 outputs each
  __shared__ float su[E_];
  su[tid]       = u[(size_t)bh * E_ + tid];
  su[tid + 256] = u[(size_t)bh * E_ + tid + 256];
  __syncthreads();
  for (int j = 0; j < 2; ++j) {
    const int ep = tid + j * 256;
    const float* wrow = Wv + ((size_t)h * E_ + ep) * E_;
    float acc = 0.0f;
    for (int d = 0; d < E_; ++d) acc += wrow[d] * su[d];
    ov[(size_t)bh * E_ + ep] = acc;
  }
}

// ---------------- kernel 7: fc + sigmoid -> out[b] -------------------------
__global__ void k_fc(const float* __restrict__ ov, const float* __restrict__ fcw,
                     const float* __restrict__ fcb, float* __restrict__ out) {
  const int b = blockIdx.x;              // B blocks
  const int tid = threadIdx.x;           // 256 threads
  const float* ob = ov + (size_t)b * H_ * E_;
  float acc = 0.0f;
  for (int i = tid; i < H_ * E_; i += 256) acc += fcw[i] * ob[i];
  __shared__ float red[256];
  red[tid] = acc;
  __syncthreads();
  for (int s = 128; s > 0; s >>= 1) { if (tid < s) red[tid] += red[tid + s]; __syncthreads(); }
  if (tid == 0) out[b] = 1.0f / (1.0f + __expf(-(red[0] + fcb[0])));
}

// ---------------------------------------------------------------------------
extern "C" void kernel_launch(void* const* d_in, const int* in_sizes, int n_in,
                              void* d_out, int out_size, void* d_ws, size_t ws_size,
                              hipStream_t stream) {
  (void)in_sizes; (void)n_in; (void)out_size; (void)ws_size;

  const int*   x     = (const int*)d_in[0];
  const float* emb   = (const float*)d_in[1];
  const float* gamma = (const float*)d_in[2];
  const float* beta  = (const float*)d_in[3];
  const float* Wq    = (const float*)d_in[4];
  const float* Wk    = (const float*)d_in[5];
  const float* Wv    = (const float*)d_in[6];
  const float* fcw   = (const float*)d_in[7];
  const float* fcb   = (const float*)d_in[8];
  float* out = (float*)d_out;

  // Workspace layout (~8.5 MiB total, fits easily in L2's 192 MB).
  char* ws = (char*)d_ws;
  size_t off = 0;
  float* e      = (float*)(ws + off); off += (size_t)B_ * S_ * E_ * 4;    // 8 MiB
  float* e_last = (float*)(ws + off); off += (size_t)B_ * E_ * 4;         // 4 KiB
  float* rpad   = (float*)(ws + off); off += (size_t)B_ * HPAD * E_ * 4;  // 64 KiB
  float* scores = (float*)(ws + off); off += (size_t)B_ * H_ * S_ * 4;    // 128 KiB
  float* wpad   = (float*)(ws + off); off += (size_t)B_ * HPAD * S_ * 4;  // 256 KiB
  float* u      = (float*)(ws + off); off += (size_t)B_ * H_ * E_ * 4;    // 32 KiB
  float* ov     = (float*)(ws + off); off += (size_t)B_ * H_ * E_ * 4;    // 32 KiB

  k_embed_ln<<<B_ * S_, 256, 0, stream>>>(x, emb, gamma, beta, e, e_last);
  k_qr      <<<B_ * HPAD, 64, 0, stream>>>(Wq, Wk, e_last, rpad);
  k_scores  <<<B_ * (S_ / 16), 32, 0, stream>>>(rpad, e, scores);
  k_weights <<<B_ * HPAD, 256, 0, stream>>>(scores, wpad);
  k_wsum    <<<B_ * (E_ / 16), 32, 0, stream>>>(wpad, e, u);
  k_ov      <<<B_ * H_, 256, 0, stream>>>(Wv, u, ov);
  k_fc      <<<B_, 256, 0, stream>>>(ov, fcw, fcb, out);
}